// KnotForward_71511205479020
// MI455X (gfx1250) — compile-verified
//
#include <hip/hip_runtime.h>

#define MAXK 128
#define HDIM 64
#define DDIM 3
#define KIN  (HDIM + DDIM)   // 67
#define KPAD 68              // K padded to multiple of 4 for 16x16x4 f32 WMMA
#define EPSF 1e-6f
#define NTHREADS 128         // 4 waves (wave32): one 16-wide N-tile per wave

typedef __attribute__((ext_vector_type(2))) float v2f;
typedef __attribute__((ext_vector_type(8))) float v8f;

// alpha = -zL / (zR - zL), valid iff |denom| > eps and alpha in (eps, 1-eps)
__device__ __forceinline__ bool xvalid(float zl, float zr, float* alpha) {
  float denom = zr - zl;
  if (!(fabsf(denom) > EPSF)) return false;
  float a = -zl / denom;
  *alpha = a;
  return (a > EPSF) && (a < 1.0f - EPSF);
}

// Segment-wise zero-crossing insertion + stable stream compaction.
// z arrays use row stride KPAD (only cols [0,HDIM) are meaningful).
__device__ int insert_crossings(const float* __restrict__ tIn,
                                const float* __restrict__ zIn,
                                float* __restrict__ tOut,
                                float* __restrict__ zOut,
                                int n, int tid,
                                int* sSegOff, float* sSlotA, int* sSlotI,
                                int* sNNew)
{
  const int nseg = n - 1;

  // 1) per-segment valid-crossing counts (sSlotI reused as scratch)
  for (int i = tid; i < nseg; i += NTHREADS) {
    const float* zl = zIn + i * KPAD;
    const float* zr = zl + KPAD;
    int cnt = 0;
    for (int h = 0; h < HDIM; ++h) {
      float a;
      if (xvalid(zl[h], zr[h], &a)) cnt++;
    }
    sSlotI[i] = cnt;
  }
  __syncthreads();

  // 2) exclusive scan over <=127 segments (trivial serial)
  if (tid == 0) {
    int acc = 0;
    for (int i = 0; i < nseg; ++i) { sSegOff[i] = acc; acc += sSlotI[i]; }
    sSegOff[nseg] = acc;
    int nn = n + acc;
    *sNNew = nn > MAXK ? MAXK : nn;
  }
  __syncthreads();
  const int nNew = *sNNew;

  // 3a) place original knots: stream pos = i + sum_{j<i} c_j
  for (int i = tid; i < n; i += NTHREADS) {
    int pos = i + sSegOff[i];
    if (pos < MAXK) { sSlotA[pos] = -1.0f; sSlotI[pos] = i; }
  }
  // 3b) place crossings with stable rank (argsort on t, tie -> hidden index)
  for (int w = tid; w < nseg * HDIM; w += NTHREADS) {
    int i = w >> 6, h = w & (HDIM - 1);
    const float* zl = zIn + i * KPAD;
    const float* zr = zl + KPAD;
    float aMine;
    if (!xvalid(zl[h], zr[h], &aMine)) continue;
    float tL = tIn[i], dt = tIn[i + 1] - tL;
    float tMine = tL + aMine * dt;
    int rank = 0;
    for (int h2 = 0; h2 < HDIM; ++h2) {
      float a2;
      if (!xvalid(zl[h2], zr[h2], &a2)) continue;
      float t2 = tL + a2 * dt;
      if ((t2 < tMine) || ((t2 == tMine) && (h2 < h))) rank++;
    }
    int pos = i + sSegOff[i] + 1 + rank;
    if (pos < MAXK) { sSlotA[pos] = aMine; sSlotI[pos] = i; }
  }
  __syncthreads();

  // 4a) fill t (tail = 1.0, matching reference)
  for (int s = tid; s < MAXK; s += NTHREADS) {
    float tv = 1.0f;
    if (s < nNew) {
      float a = sSlotA[s]; int i = sSlotI[s];
      tv = (a < 0.0f) ? tIn[i] : tIn[i] + a * (tIn[i + 1] - tIn[i]);
    }
    tOut[s] = tv;
  }
  // 4b) fill z (interpolated at crossings, copied at originals, 0 on tail)
  for (int e = tid; e < MAXK * HDIM; e += NTHREADS) {
    int s = e >> 6, h = e & (HDIM - 1);
    float zv = 0.0f;
    if (s < nNew) {
      float a = sSlotA[s]; int i = sSlotI[s];
      float zl = zIn[i * KPAD + h];
      zv = (a < 0.0f) ? zl : zl + a * (zIn[(i + 1) * KPAD + h] - zl);
    }
    zOut[s * KPAD + h] = zv;
  }
  __syncthreads();
  return nNew;
}

// [MAXK x KPAD] @ [KPAD x HDIM] + bias using native fp32 WMMA 16x16x4.
// Wave w owns N-tile w (HDIM/16 == 4 waves); loops the 8 M-tiles; K = 17 steps.
__device__ __forceinline__ void wmma_gemm(const float* __restrict__ sHin,
                                          const float* __restrict__ sW,
                                          const float* __restrict__ sBias,
                                          float* __restrict__ zOut, int tid)
{
  const int wave = tid >> 5;
  const int lane = tid & 31;
  const int lr   = lane & 15;   // row/col within tile
  const int lh   = lane >> 4;   // K-pair selector (A/B), M-half selector (C/D)
  const int n0   = wave * 16;
  const float bias = sBias[n0 + lr];
  for (int mt = 0; mt < MAXK / 16; ++mt) {
    const int m0 = mt * 16;
    v8f c;
#pragma unroll
    for (int r = 0; r < 8; ++r) c[r] = bias;
#pragma unroll
    for (int kt = 0; kt < KPAD / 4; ++kt) {
      const int k = kt * 4 + 2 * lh;
      v2f a, w;
      a.x = sHin[(m0 + lr) * KPAD + k];       // A: 16x4 tile, row m0+lr
      a.y = sHin[(m0 + lr) * KPAD + k + 1];
      w.x = sW[(n0 + lr) * KPAD + k];         // B = W^T: col n0+lr of output
      w.y = sW[(n0 + lr) * KPAD + k + 1];
      c = __builtin_amdgcn_wmma_f32_16x16x4_f32(false, a, false, w,
                                                (short)0, c, false, false);
    }
#pragma unroll
    for (int r = 0; r < 8; ++r)               // C/D: row r+8*lh, col lr
      zOut[(m0 + r + 8 * lh) * KPAD + (n0 + lr)] = c[r];
  }
}

__global__ __launch_bounds__(NTHREADS)
void knot_forward_kernel(const float* __restrict__ end_pts,
                         const float* __restrict__ start_pts,
                         const float* __restrict__ W0, const float* __restrict__ b0,
                         const float* __restrict__ W1, const float* __restrict__ b1,
                         const float* __restrict__ W2, const float* __restrict__ b2,
                         const float* __restrict__ W3, const float* __restrict__ b3,
                         float* __restrict__ out_t, float* __restrict__ out_v,
                         float* __restrict__ out_y)
{
  __shared__ float bufZ[2][MAXK * KPAD];   // ping-pong z; free one aliases h_in
  __shared__ float bufT[2][MAXK];
  __shared__ float sW[HDIM * KPAD];
  __shared__ float sBias[HDIM];
  __shared__ float sW3[HDIM];
  __shared__ float sSP[DDIM];
  __shared__ float sDV[DDIM];
  __shared__ int   sSegOff[MAXK + 1];
  __shared__ float sSlotA[MAXK];
  __shared__ int   sSlotI[MAXK];
  __shared__ int   sNNew;

  const int b   = blockIdx.x;
  const int tid = threadIdx.x;

  if (tid < DDIM) {
    float sp = start_pts[b * DDIM + tid];
    sSP[tid] = sp;
    sDV[tid] = end_pts[b * DDIM + tid] - sp;
  }
  if (tid < HDIM) sW3[tid] = W3[tid];
  __syncthreads();

  // ---- layer 0: z(t) = (d@W0^T)*t + (s@W0^T + b0), knots t = {0, 1} ----
  if (tid < HDIM) {
    float a0 = 0.0f, be = b0[tid];
#pragma unroll
    for (int d = 0; d < DDIM; ++d) {
      float w = W0[tid * DDIM + d];
      a0 += sDV[d] * w;
      be += sSP[d] * w;
    }
    bufZ[0][0 * KPAD + tid] = be;
    bufZ[0][1 * KPAD + tid] = a0 + be;
  }
  if (tid == 0) { bufT[0][0] = 0.0f; bufT[0][1] = 1.0f; }
  __syncthreads();

  int n = insert_crossings(bufT[0], bufZ[0], bufT[1], bufZ[1], 2, tid,
                           sSegOff, sSlotA, sSlotI, &sNNew);
  int cur = 1;

  // ---- hidden layers 1 and 2 ----
  for (int L = 0; L < 2; ++L) {
    const float* Wl = (L == 0) ? W1 : W2;
    const float* bl = (L == 0) ? b1 : b2;
    // stage W^T source [HDIM x KIN] into LDS, pad col 67 with 0
    for (int e = tid; e < HDIM * KPAD; e += NTHREADS) {
      int r = e / KPAD, c2 = e - r * KPAD;
      sW[e] = (c2 < KIN) ? Wl[r * KIN + c2] : 0.0f;
    }
    if (tid < HDIM) sBias[tid] = bl[tid];
    // h_in = [relu(z), start + t*d, 0-pad]; invalid rows zeroed. Built into
    // the free z buffer (its data is dead), then GEMM overwrites bufZ[cur].
    float* hin      = bufZ[cur ^ 1];
    const float* zc = bufZ[cur];
    const float* tc = bufT[cur];
    for (int e = tid; e < MAXK * KPAD; e += NTHREADS) {
      int k = e / KPAD, j = e - k * KPAD;
      float v = 0.0f;
      if (k < n) {
        if (j < HDIM)       { float z = zc[k * KPAD + j]; v = z > 0.0f ? z : 0.0f; }
        else if (j < KIN)   { int dd = j - HDIM; v = sSP[dd] + tc[k] * sDV[dd]; }
      }
      hin[e] = v;
    }
    __syncthreads();
    wmma_gemm(hin, sW, sBias, bufZ[cur], tid);
    __syncthreads();
    n = insert_crossings(bufT[cur], bufZ[cur], bufT[cur ^ 1], bufZ[cur ^ 1],
                         n, tid, sSegOff, sSlotA, sSlotI, &sNNew);
    cur ^= 1;
  }

  // ---- output head (64 -> 1) and final writes ----
  const float b3v = b3[0];
  {
    const float* zc = bufZ[cur];
    const float* tc = bufT[cur];
    float y = 0.0f;
    int vv = (tid < n) ? 1 : 0;
    if (vv) {
      for (int h = 0; h < HDIM; ++h) {
        float z = zc[tid * KPAD + h];
        y = fmaf(z > 0.0f ? z : 0.0f, sW3[h], y);
      }
      y += b3v;
    }
    out_t[b * MAXK + tid] = tc[tid];
    out_v[b * MAXK + tid] = (float)vv;
    out_y[b * MAXK + tid] = y;
  }
}

extern "C" void kernel_launch(void* const* d_in, const int* in_sizes, int n_in,
                              void* d_out, int out_size, void* d_ws, size_t ws_size,
                              hipStream_t stream) {
  const float* end_pts   = (const float*)d_in[0];
  const float* start_pts = (const float*)d_in[1];
  const float* W0 = (const float*)d_in[2];
  const float* b0 = (const float*)d_in[3];
  const float* W1 = (const float*)d_in[4];
  const float* b1 = (const float*)d_in[5];
  const float* W2 = (const float*)d_in[6];
  const float* b2 = (const float*)d_in[7];
  const float* W3 = (const float*)d_in[8];
  const float* b3 = (const float*)d_in[9];

  const int B = in_sizes[0] / DDIM;   // 256
  float* out   = (float*)d_out;       // [t | valid | y], each B*MAXK floats
  float* out_t = out;
  float* out_v = out + (size_t)B * MAXK;
  float* out_y = out + (size_t)2 * B * MAXK;

  knot_forward_kernel<<<dim3(B), dim3(NTHREADS), 0, stream>>>(
      end_pts, start_pts, W0, b0, W1, b1, W2, b2, W3, b3,
      out_t, out_v, out_y);
}